// TorchSOM_81707457839333
// MI455X (gfx1250) — compile-verified
//
#include <hip/hip_runtime.h>
#include <cstdint>

// SOM step for X=Y=128, F=256, B=2048, SIGMA=2, LR=0.5 on gfx1250 (wave32, WMMA).
#define XDIM 128
#define YDIM 128
#define FDIM 256
#define BDIM 2048
#define NNEUR (XDIM * YDIM)
#define INV2S2 0.125f        // 1/(2*SIGMA^2)
#define LRV 0.5f

typedef __attribute__((ext_vector_type(2))) float v2f;
typedef __attribute__((ext_vector_type(8))) float v8f;

__device__ __forceinline__ unsigned int fkey(float f) {
  // order-preserving float -> u32 (handles negatives)
  unsigned int b = __float_as_uint(f);
  return (b & 0x80000000u) ? ~b : (b | 0x80000000u);
}

__device__ __forceinline__ v2f ldfrag(const float* p) {
  float2 t = *(const float2*)p;
  v2f r; r.x = t.x; r.y = t.y;
  return r;
}

// ---------------- init: bmu_packed = +inf ----------------
__global__ void som_init(unsigned long long* __restrict__ bmu) {
  int b = blockIdx.x * blockDim.x + threadIdx.x;
  if (b < BDIM) bmu[b] = 0xFFFFFFFFFFFFFFFFull;
}

// ---------------- ||w_n||^2 per neuron ----------------
__global__ __launch_bounds__(256) void som_wnorm(const float* __restrict__ w,
                                                 float* __restrict__ wnorm2) {
  int n = blockIdx.x * blockDim.x + threadIdx.x;  // 16384 threads
  const float4* p = (const float4*)(w + (size_t)n * FDIM);
  float s = 0.f;
#pragma unroll 8
  for (int i = 0; i < FDIM / 4; ++i) {
    float4 v = p[i];
    s = fmaf(v.x, v.x, s); s = fmaf(v.y, v.y, s);
    s = fmaf(v.z, v.z, s); s = fmaf(v.w, v.w, s);
  }
  wnorm2[n] = s;
}

// ---------------- BMU search: fused GEMM(data @ W^T) + argmin, 2x2 blocked --------
// wave = 32 samples (2 M-tiles) x 1024-neuron chunk iterated as 32 N-tile pairs.
// Per k-step: 2 A-frag + 2 B-frag loads feed 4 WMMAs (1:1 load:wmma, 8 FLOP/B).
__global__ __launch_bounds__(256) void som_bmu_gemm(const float* __restrict__ data,
                                                    const float* __restrict__ w,
                                                    const float* __restrict__ wnorm2,
                                                    unsigned long long* __restrict__ bmu) {
  const int wave = (blockIdx.x * blockDim.x + threadIdx.x) >> 5;  // 0..1023
  const int lane = threadIdx.x & 31;
  const int half = lane >> 4;
  const int l16  = lane & 15;
  const int b0    = (wave >> 4) << 5;    // sample base (32 samples)
  const int nbase = (wave & 15) << 10;   // 1024-neuron chunk

  float        minv0[8], minv1[8];
  unsigned int mini0[8], mini1[8];
#pragma unroll
  for (int r = 0; r < 8; ++r) {
    minv0[r] = 3.4e38f; mini0[r] = 0u;
    minv1[r] = 3.4e38f; mini1[r] = 0u;
  }

  const float* a0row = data + (size_t)(b0 + l16) * FDIM;        // samples b0..b0+15
  const float* a1row = data + (size_t)(b0 + 16 + l16) * FDIM;   // samples b0+16..b0+31

  for (int t = 0; t < 32; ++t) {
    const int n0 = nbase + (t << 5);
    const int n1 = n0 + 16;
    const float* b0row = w + (size_t)(n0 + l16) * FDIM;
    const float* b1row = w + (size_t)(n1 + l16) * FDIM;
    v8f c00 = {0.f,0.f,0.f,0.f,0.f,0.f,0.f,0.f};
    v8f c01 = c00, c10 = c00, c11 = c00;
#pragma unroll 2
    for (int k = 0; k < FDIM; k += 4) {
      const int kk = k + (half << 1);
      v2f a0 = ldfrag(a0row + kk);
      v2f a1 = ldfrag(a1row + kk);
      v2f bb0 = ldfrag(b0row + kk);
      v2f bb1 = ldfrag(b1row + kk);
      c00 = __builtin_amdgcn_wmma_f32_16x16x4_f32(false, a0, false, bb0, (short)0, c00, false, false);
      c01 = __builtin_amdgcn_wmma_f32_16x16x4_f32(false, a0, false, bb1, (short)0, c01, false, false);
      c10 = __builtin_amdgcn_wmma_f32_16x16x4_f32(false, a1, false, bb0, (short)0, c10, false, false);
      c11 = __builtin_amdgcn_wmma_f32_16x16x4_f32(false, a1, false, bb1, (short)0, c11, false, false);
    }
    // epilogue: d2 = ||w_n||^2 - 2*score (sample norm constant -> argmin invariant)
    const float wn0 = wnorm2[n0 + l16];
    const float wn1 = wnorm2[n1 + l16];
    const unsigned int ni0 = (unsigned int)(n0 + l16);
    const unsigned int ni1 = (unsigned int)(n1 + l16);
#pragma unroll
    for (int r = 0; r < 8; ++r) {
      float d;
      d = fmaf(-2.0f, c00[r], wn0);
      if (d < minv0[r] || (d == minv0[r] && ni0 < mini0[r])) { minv0[r] = d; mini0[r] = ni0; }
      d = fmaf(-2.0f, c01[r], wn1);
      if (d < minv0[r] || (d == minv0[r] && ni1 < mini0[r])) { minv0[r] = d; mini0[r] = ni1; }
      d = fmaf(-2.0f, c10[r], wn0);
      if (d < minv1[r] || (d == minv1[r] && ni0 < mini1[r])) { minv1[r] = d; mini1[r] = ni0; }
      d = fmaf(-2.0f, c11[r], wn1);
      if (d < minv1[r] || (d == minv1[r] && ni1 < mini1[r])) { minv1[r] = d; mini1[r] = ni1; }
    }
  }
  // reduce across the 16 lanes of each wave half (C layout keeps halves independent)
#pragma unroll
  for (int r = 0; r < 8; ++r) {
#pragma unroll
    for (int m = 8; m >= 1; m >>= 1) {
      float ov; unsigned int oi;
      ov = __shfl_xor(minv0[r], m, 16); oi = __shfl_xor(mini0[r], m, 16);
      if (ov < minv0[r] || (ov == minv0[r] && oi < mini0[r])) { minv0[r] = ov; mini0[r] = oi; }
      ov = __shfl_xor(minv1[r], m, 16); oi = __shfl_xor(mini1[r], m, 16);
      if (ov < minv1[r] || (ov == minv1[r] && oi < mini1[r])) { minv1[r] = ov; mini1[r] = oi; }
    }
  }
  if (l16 == 0) {
#pragma unroll
    for (int r = 0; r < 8; ++r) {
      const int s = r + (half << 3);
      unsigned long long p0 =
          ((unsigned long long)fkey(minv0[r]) << 32) | (unsigned long long)mini0[r];
      unsigned long long p1 =
          ((unsigned long long)fkey(minv1[r]) << 32) | (unsigned long long)mini1[r];
      atomicMin(&bmu[b0 + s], p0);
      atomicMin(&bmu[b0 + 16 + s], p1);
    }
  }
}

// ---------------- neighborhood tables gx[b,x], gy[b,y] ----------------
__global__ void som_gtab(const unsigned long long* __restrict__ bmu,
                         float* __restrict__ gx, float* __restrict__ gy) {
  int tid = blockIdx.x * blockDim.x + threadIdx.x;
  const int total = BDIM * XDIM;
  if (tid < total) {
    const int b = tid >> 7, x = tid & 127;
    const unsigned int idx = (unsigned int)(bmu[b] & 0xFFFFFFFFu);
    const float d = (float)x - (float)(idx >> 7);   // bmu_x = idx / Y
    gx[tid] = expf(-d * d * INV2S2);
  } else {
    tid -= total;
    const int b = tid >> 7, y = tid & 127;
    const unsigned int idx = (unsigned int)(bmu[b] & 0xFFFFFFFFu);
    const float d = (float)y - (float)(idx & 127u); // bmu_y = idx % Y
    gy[tid] = expf(-d * d * INV2S2);
  }
}

// ---------------- nb_sum[x,y] = sum_b gx[b,x]*gy[b,y] ----------------
__global__ __launch_bounds__(256) void som_nbsum(const float* __restrict__ gx,
                                                 const float* __restrict__ gy,
                                                 float* __restrict__ nbsum) {
  const int n = blockIdx.x * blockDim.x + threadIdx.x;  // 16384
  const int x = n >> 7, y = n & 127;
  float s = 0.f;
  for (int b = 0; b < BDIM; ++b)
    s = fmaf(gx[b * XDIM + x], gy[b * YDIM + y], s);
  nbsum[n] = s;
}

// ---------------- weighted_data GEMM (nb^T @ data) + fused update, 2x2 blocked -----
// wave -> 32 neurons (2 M-tiles) x 32 features (2 N-tiles), K = B = 2048 step 4.
// nb synthesized on the fly: nb[b, n] = gx[b, x0] * gy[b, y0 + m].
__global__ __launch_bounds__(256) void som_update_gemm(const float* __restrict__ data,
                                                       const float* __restrict__ w,
                                                       const float* __restrict__ gx,
                                                       const float* __restrict__ gy,
                                                       const float* __restrict__ nbsum,
                                                       float* __restrict__ out) {
  const int wave = (blockIdx.x * blockDim.x + threadIdx.x) >> 5;  // 0..4095
  const int lane = threadIdx.x & 31;
  const int half = lane >> 4;
  const int l16  = lane & 15;
  const int mt = wave >> 3;            // neuron tile-pair 0..511
  const int ft = wave & 7;             // feature tile-pair 0..7
  const int n0 = mt << 5;              // 32 neurons, same grid row (32 | 128)
  const int x0 = n0 >> 7;
  const int y0 = n0 & 127;
  const int f0 = ft << 5;              // 32 features

  v8f c00 = {0.f,0.f,0.f,0.f,0.f,0.f,0.f,0.f};
  v8f c01 = c00, c10 = c00, c11 = c00;
  const float* gxp  = gx + x0;              // uniform per wave -> scalar path
  const float* gyp0 = gy + y0 + l16;        // neuron rows n0..n0+15
  const float* gyp1 = gy + y0 + 16 + l16;   // neuron rows n0+16..n0+31
  const float* dp0  = data + f0 + l16;      // feature cols f0..f0+15
  const float* dp1  = data + f0 + 16 + l16; // feature cols f0+16..f0+31

#pragma unroll 2
  for (int k = 0; k < BDIM; k += 4) {
    const int kk = k + (half << 1);
    const float gxv0 = gxp[kk * XDIM];
    const float gxv1 = gxp[(kk + 1) * XDIM];
    v2f a0, a1, b0, b1;
    a0.x = gxv0 * gyp0[kk * YDIM];
    a0.y = gxv1 * gyp0[(kk + 1) * YDIM];
    a1.x = gxv0 * gyp1[kk * YDIM];
    a1.y = gxv1 * gyp1[(kk + 1) * YDIM];
    b0.x = dp0[kk * FDIM];
    b0.y = dp0[(kk + 1) * FDIM];
    b1.x = dp1[kk * FDIM];
    b1.y = dp1[(kk + 1) * FDIM];
    c00 = __builtin_amdgcn_wmma_f32_16x16x4_f32(false, a0, false, b0, (short)0, c00, false, false);
    c01 = __builtin_amdgcn_wmma_f32_16x16x4_f32(false, a0, false, b1, (short)0, c01, false, false);
    c10 = __builtin_amdgcn_wmma_f32_16x16x4_f32(false, a1, false, b0, (short)0, c10, false, false);
    c11 = __builtin_amdgcn_wmma_f32_16x16x4_f32(false, a1, false, b1, (short)0, c11, false, false);
  }

  // fused: new_w = w + (LR/B) * (weighted_data - nb_sum[n] * w)
  const float scale = LRV / (float)BDIM;
  const int fA = f0 + l16;
  const int fB = f0 + 16 + l16;
#pragma unroll
  for (int r = 0; r < 8; ++r) {
    const int nA = n0 + r + (half << 3);
    const int nB = nA + 16;
    const float nsA = nbsum[nA];
    const float nsB = nbsum[nB];
    float wv;
    wv = w[(size_t)nA * FDIM + fA];
    out[(size_t)nA * FDIM + fA] = wv + scale * (c00[r] - nsA * wv);
    wv = w[(size_t)nA * FDIM + fB];
    out[(size_t)nA * FDIM + fB] = wv + scale * (c01[r] - nsA * wv);
    wv = w[(size_t)nB * FDIM + fA];
    out[(size_t)nB * FDIM + fA] = wv + scale * (c10[r] - nsB * wv);
    wv = w[(size_t)nB * FDIM + fB];
    out[(size_t)nB * FDIM + fB] = wv + scale * (c11[r] - nsB * wv);
  }
}

extern "C" void kernel_launch(void* const* d_in, const int* in_sizes, int n_in,
                              void* d_out, int out_size, void* d_ws, size_t ws_size,
                              hipStream_t stream) {
  const float* data = (const float*)d_in[0];   // [B, F]
  const float* w    = (const float*)d_in[1];   // [X, Y, F] = [N, F]
  float* out        = (float*)d_out;           // [N, F]

  char* ws = (char*)d_ws;
  unsigned long long* bmu = (unsigned long long*)(ws);                 // 16 KB
  float* wnorm2 = (float*)(ws + 16384);                                // 64 KB
  float* gx     = (float*)(ws + 16384 + 65536);                        // 1 MB
  float* gy     = (float*)(ws + 16384 + 65536 + 1048576);              // 1 MB
  float* nbsum  = (float*)(ws + 16384 + 65536 + 2 * 1048576);          // 64 KB

  som_init<<<(BDIM + 255) / 256, 256, 0, stream>>>(bmu);
  som_wnorm<<<NNEUR / 256, 256, 0, stream>>>(w, wnorm2);
  // 1024 waves: 64 sample-pairs x 16 neuron-chunks -> 128 blocks of 8 waves
  som_bmu_gemm<<<128, 256, 0, stream>>>(data, w, wnorm2, bmu);
  som_gtab<<<(2 * BDIM * XDIM) / 256, 256, 0, stream>>>(bmu, gx, gy);
  som_nbsum<<<NNEUR / 256, 256, 0, stream>>>(gx, gy, nbsum);
  // 4096 waves: 512 neuron-tile-pairs x 8 feature-tile-pairs -> 512 blocks of 8 waves
  som_update_gemm<<<512, 256, 0, stream>>>(data, w, gx, gy, nbsum, out);
}